// GbufferCrossBilateral_43078521979262
// MI455X (gfx1250) — compile-verified
//
#include <hip/hip_runtime.h>
#include <cstdint>

// Problem constants (match reference).
#define B_      4
#define C_      3
#define H_      720
#define W_      1280
#define WIN_    11
#define PAD_    5

// Tiling: 32x16 output tile, 42x26 halo.
#define TILE_W  32
#define TILE_H  16
#define HALO_W  (TILE_W + WIN_ - 1)   // 42
#define HALO_H  (TILE_H + WIN_ - 1)   // 26
#define NPIX    (HALO_H * HALO_W)     // 1092
#define STRIDE  12                    // floats per halo pixel (48 B, 16B-aligned)
#define THREADS 256

// exp(-(dA/0.01 + dN/0.1 + dD/0.1)) = exp2(KA*dA + KN*dN + KD*dD)
#define KA (-144.26950408889634f)   // -100 * log2(e)
#define KN (-14.426950408889634f)   // -10  * log2(e)
#define KD (-14.426950408889634f)

typedef __attribute__((address_space(1))) int g_i32;   // global b32
typedef __attribute__((address_space(3))) int l_i32;   // LDS b32
typedef __attribute__((ext_vector_type(4))) float v4f;

__device__ __forceinline__ void async_load_f32_to_lds(const float* g, const float* l) {
#if defined(__has_builtin) && __has_builtin(__builtin_amdgcn_global_load_async_to_lds_b32)
  __builtin_amdgcn_global_load_async_to_lds_b32(
      (g_i32*)(unsigned long long)(uintptr_t)g,
      (l_i32*)(unsigned)(uintptr_t)l, 0, 0);
#else
  unsigned loff = (unsigned)(uintptr_t)l;
  asm volatile("global_load_async_to_lds_b32 %0, %1, off"
               :: "v"(loff), "v"(g) : "memory");
#endif
}

__device__ __forceinline__ void wait_async_zero() {
#if defined(__has_builtin) && __has_builtin(__builtin_amdgcn_s_wait_asynccnt)
  __builtin_amdgcn_s_wait_asynccnt(0);
#else
  asm volatile("s_wait_asynccnt 0" ::: "memory");
#endif
}

__global__ __launch_bounds__(THREADS)
void GbufferCrossBilateral_43078521979262_kernel(
    const float* __restrict__ inp, const float* __restrict__ alb,
    const float* __restrict__ nrm, const float* __restrict__ dep,
    float* __restrict__ out)
{
  // Interleaved halo: [a0 a1 a2 n0 | n1 n2 d S | i0 i1 i2 pad] per pixel.
  __shared__ float smem[NPIX * STRIDE];   // 52,416 B

  const int tid = threadIdx.x;
  const int bx0 = blockIdx.x * TILE_W;
  const int by0 = blockIdx.y * TILE_H;
  const int bb  = blockIdx.z;
  const size_t HW = (size_t)H_ * W_;

  // Sources for struct slots 0..6 (guides) and 8..10 (input). Slot 7 = S(q).
  const float* srcs[10];
  srcs[0] = alb + (size_t)(bb * 3 + 0) * HW;
  srcs[1] = alb + (size_t)(bb * 3 + 1) * HW;
  srcs[2] = alb + (size_t)(bb * 3 + 2) * HW;
  srcs[3] = nrm + (size_t)(bb * 3 + 0) * HW;
  srcs[4] = nrm + (size_t)(bb * 3 + 1) * HW;
  srcs[5] = nrm + (size_t)(bb * 3 + 2) * HW;
  srcs[6] = dep + (size_t)bb * HW;
  srcs[7] = inp + (size_t)(bb * 3 + 0) * HW;   // -> slot 8
  srcs[8] = inp + (size_t)(bb * 3 + 1) * HW;   // -> slot 9
  srcs[9] = inp + (size_t)(bb * 3 + 2) * HW;   // -> slot 10

  // ---- Pass 1: async global->LDS staging, edge-clamped per lane. ----
  for (int i = tid; i < NPIX; i += THREADS) {
    const int hy = i / HALO_W;
    const int hx = i - hy * HALO_W;
    int gy = by0 - PAD_ + hy; gy = gy < 0 ? 0 : (gy > H_ - 1 ? H_ - 1 : gy);
    int gx = bx0 - PAD_ + hx; gx = gx < 0 ? 0 : (gx > W_ - 1 ? W_ - 1 : gx);
    const size_t goff = (size_t)gy * W_ + gx;
    const int base = i * STRIDE;
#pragma unroll
    for (int p = 0; p < 7; ++p)
      async_load_f32_to_lds(srcs[p] + goff, &smem[base + p]);
#pragma unroll
    for (int p = 7; p < 10; ++p)
      async_load_f32_to_lds(srcs[p] + goff, &smem[base + p + 1]);
  }
  wait_async_zero();
  __syncthreads();

  // ---- Pass 2: per-halo-pixel S(q) = KA*|a|^2 + KN*|n|^2 + KD*d^2. ----
  for (int i = tid; i < NPIX; i += THREADS) {
    const int base = i * STRIDE;
    const v4f q0 = *(const v4f*)&smem[base];       // a0 a1 a2 n0
    const v4f q1 = *(const v4f*)&smem[base + 4];   // n1 n2 d  --
    float s = q0.x * q0.x;
    s = __builtin_fmaf(q0.y, q0.y, s);
    s = __builtin_fmaf(q0.z, q0.z, s);
    s *= KA;
    float sn = q0.w * q0.w;
    sn = __builtin_fmaf(q1.x, q1.x, sn);
    sn = __builtin_fmaf(q1.y, q1.y, sn);
    s = __builtin_fmaf(sn, KN, s);
    s = __builtin_fmaf(q1.z * q1.z, KD, s);
    smem[base + 7] = s;
  }
  __syncthreads();

  // ---- Compute: 2 output rows per thread, 121 taps each. ----
  const int tx  = tid & (TILE_W - 1);
  const int ty0 = tid >> 5;

#pragma unroll
  for (int sub = 0; sub < 2; ++sub) {
    const int py = ty0 + sub * 8;
    const int cbase = (((py + PAD_) * HALO_W) + (tx + PAD_)) * STRIDE;
    const v4f c0 = *(const v4f*)&smem[cbase];       // a0 a1 a2 n0
    const v4f c1 = *(const v4f*)&smem[cbase + 4];   // n1 n2 d  S(p)
    const float sp = c1.w;
    const float f0 = -2.0f * KA * c0.x;
    const float f1 = -2.0f * KA * c0.y;
    const float f2 = -2.0f * KA * c0.z;
    const float f3 = -2.0f * KN * c0.w;
    const float f4 = -2.0f * KN * c1.x;
    const float f5 = -2.0f * KN * c1.y;
    const float f6 = -2.0f * KD * c1.z;

    float acc0 = 0.f, acc1 = 0.f, acc2 = 0.f, ws = 0.f;

    for (int dy = 0; dy < WIN_; ++dy) {
      const int rbase = ((py + dy) * HALO_W + tx) * STRIDE;
#pragma unroll
      for (int dx = 0; dx < WIN_; ++dx) {
        const int hb = rbase + dx * STRIDE;
        const v4f g0 = *(const v4f*)&smem[hb];        // a0 a1 a2 n0
        const v4f g1 = *(const v4f*)&smem[hb + 4];    // n1 n2 d  S(q)
        const v4f gi = *(const v4f*)&smem[hb + 8];    // i0 i1 i2 --
        float t = sp + g1.w;                          // S(p)+S(q)
        t = __builtin_fmaf(f0, g0.x, t);
        t = __builtin_fmaf(f1, g0.y, t);
        t = __builtin_fmaf(f2, g0.z, t);
        t = __builtin_fmaf(f3, g0.w, t);
        t = __builtin_fmaf(f4, g1.x, t);
        t = __builtin_fmaf(f5, g1.y, t);
        t = __builtin_fmaf(f6, g1.z, t);
        const float w = __builtin_amdgcn_exp2f(t);    // v_exp_f32, 1/tap
        acc0 = __builtin_fmaf(w, gi.x, acc0);
        acc1 = __builtin_fmaf(w, gi.y, acc1);
        acc2 = __builtin_fmaf(w, gi.z, acc2);
        ws  += w;
      }
    }

    const float inv = __builtin_amdgcn_rcpf(fmaxf(ws, 1e-10f));
    const int oy = by0 + py, ox = bx0 + tx;
    const size_t o = (size_t)(bb * 3) * HW + (size_t)oy * W_ + ox;
    out[o]          = acc0 * inv;
    out[o + HW]     = acc1 * inv;
    out[o + 2 * HW] = acc2 * inv;
  }
}

extern "C" void kernel_launch(void* const* d_in, const int* in_sizes, int n_in,
                              void* d_out, int out_size, void* d_ws, size_t ws_size,
                              hipStream_t stream) {
  const float* inp = (const float*)d_in[0];
  const float* alb = (const float*)d_in[1];
  const float* nrm = (const float*)d_in[2];
  const float* dep = (const float*)d_in[3];
  // d_in[4] = win_size (scalar, always 11) — compiled in as WIN_.
  float* out = (float*)d_out;

  dim3 grid(W_ / TILE_W, H_ / TILE_H, B_);   // 40 x 45 x 4, exact tiling
  hipLaunchKernelGGL(GbufferCrossBilateral_43078521979262_kernel,
                     grid, dim3(THREADS), 0, stream, inp, alb, nrm, dep, out);
}